// DynamicFilter_72043781423713
// MI455X (gfx1250) — compile-verified
//
#include <hip/hip_runtime.h>
#include <hip/hip_bf16.h>
#include <math.h>

#define N_    4
#define C_    32
#define H_    256
#define W_    256
#define G_    8
#define K2_   9
#define GK2_  72          // G * K^2
#define HW_   (H_ * W_)
#define EPS_  1e-5f
#define ROWS  32          // rows per stencil band

typedef __attribute__((ext_vector_type(2))) float v2f;
typedef __attribute__((ext_vector_type(8))) float v8f;

// ---------------------------------------------------------------------------
// Kernel 1: per-(n,c) mean over H*W.  128 blocks x 256 threads, float4 loads.
// ---------------------------------------------------------------------------
__global__ __launch_bounds__(256)
void pool_mean_kernel(const float* __restrict__ x, float* __restrict__ pooled) {
    const int nc  = blockIdx.x;                    // 0..127  (n*32 + c)
    const int tid = threadIdx.x;                   // 0..255
    const float4* base = (const float4*)(x + (size_t)nc * HW_);
    float s = 0.0f;
    #pragma unroll 4
    for (int i = tid; i < HW_ / 4; i += 256) {     // 64 coalesced float4s / thread
        float4 v = base[i];
        s += v.x + v.y + v.z + v.w;
    }
    __shared__ float red[256];
    red[tid] = s;
    __syncthreads();
    for (int off = 128; off > 0; off >>= 1) {
        if (tid < off) red[tid] += red[tid + off];
        __syncthreads();
    }
    if (tid == 0) pooled[nc] = red[0] * (1.0f / (float)HW_);
}

// ---------------------------------------------------------------------------
// Kernel 2: filt = tanh(BN(pooled @ conv_w.T)) via V_WMMA_F32_16X16X4_F32.
// One wave32. M: 4 rows padded to 16. K=32 as 8 steps of 4. N=72 as 5 tiles.
// A layout (16x4 f32): lanes 0-15 hold M=lane, vgpr pair = K {0,1};
//                      lanes 16-31 hold M=lane-16, K {2,3}.
// B layout (4x16 f32): lanes 0-15 hold N=lane, vgpr pair = K {0,1};
//                      lanes 16-31 hold N=lane-16, K {2,3}.
// D layout: vgpr r, lanes 0-15 -> (M=r, N=lane). We need only M=0..3.
// ---------------------------------------------------------------------------
__global__ __launch_bounds__(32)
void filter_wmma_kernel(const float* __restrict__ pooled,
                        const float* __restrict__ conv_w,
                        const float* __restrict__ bn_gamma,
                        const float* __restrict__ bn_beta,
                        const float* __restrict__ bn_mean,
                        const float* __restrict__ bn_var,
                        float* __restrict__ filt) {
    const int lane  = threadIdx.x;        // 0..31
    const int mn    = lane & 15;          // M index for A, N index for B/D
    const int khalf = (lane >> 4) * 2;    // 0 or 2: K sub-pair within the 4

    const int mclamp = (mn < N_) ? mn : 0;          // safe row for A loads
    const float mkeep = (mn < N_) ? 1.0f : 0.0f;    // zero the padded rows

    for (int ntile = 0; ntile < 5; ++ntile) {       // N = 5*16 = 80 >= 72
        const int j      = ntile * 16 + mn;         // output column
        const int jclamp = (j < GK2_) ? j : (GK2_ - 1);
        const float jkeep = (j < GK2_) ? 1.0f : 0.0f;

        v8f acc = {};
        #pragma unroll
        for (int kb = 0; kb < C_; kb += 4) {        // 8 WMMA k-steps
            v2f a, b;
            a.x = mkeep * pooled[mclamp * C_ + kb + khalf];
            a.y = mkeep * pooled[mclamp * C_ + kb + khalf + 1];
            // f[n,j] = sum_k pooled[n,k] * conv_w[j,k]  =>  B[k][j] = conv_w[j*C_+k]
            b.x = jkeep * conv_w[jclamp * C_ + kb + khalf];
            b.y = jkeep * conv_w[jclamp * C_ + kb + khalf + 1];
            acc = __builtin_amdgcn_wmma_f32_16x16x4_f32(
                false, a, false, b, (short)0, acc, false, false);
        }

        if (lane < 16 && j < GK2_) {
            const float sc = bn_gamma[j] * rsqrtf(bn_var[j] + EPS_);
            const float mu = bn_mean[j];
            const float bt = bn_beta[j];
            #pragma unroll
            for (int r = 0; r < N_; ++r) {          // rows M=0..3 live in acc[0..3]
                float f = (acc[r] - mu) * sc + bt;
                filt[r * GK2_ + j] = tanhf(f);
            }
        }
    }
}

// ---------------------------------------------------------------------------
// Kernel 3: HBM-bound 3x3 reflect-padded dynamic depthwise stencil + epilogue.
// Block = one (n,c) plane band of ROWS rows; LDS-stages ROWS+2 rows (~6% halo
// overfetch); thread = one column; coalesced b32 loads/stores.
// ---------------------------------------------------------------------------
__global__ __launch_bounds__(256)
void dynfilt_stencil_kernel(const float* __restrict__ x,
                            const float* __restrict__ pooled,
                            const float* __restrict__ filt,
                            const float* __restrict__ lamb_l,
                            const float* __restrict__ lamb_h,
                            const float* __restrict__ inside_all,
                            float* __restrict__ out) {
    __shared__ float tile[ROWS + 2][W_];            // 34 KB

    const int tid  = threadIdx.x;                   // column 0..255
    const int bid  = blockIdx.x;
    const int band = bid & (H_ / ROWS - 1);         // 0..7
    const int nc   = bid >> 3;                      // n*32 + c
    const int n    = nc >> 5;
    const int c    = nc & 31;
    const int g    = c >> 2;                        // group = c / (C/G)

    const float* base  = x   + (size_t)nc * HW_;
    float*       obase = out + (size_t)nc * HW_;

    // Stage ROWS+2 rows with row reflection (p=1): -1 -> 1, 256 -> 254.
    for (int r = 0; r < ROWS + 2; ++r) {
        int gr = band * ROWS + r - 1;
        gr = (gr < 0) ? -gr : gr;
        gr = (gr > H_ - 1) ? (2 * (H_ - 1) - gr) : gr;
        tile[r][tid] = base[gr * W_ + tid];
    }
    __syncthreads();

    // Uniform per-block scalars.
    float f[K2_];
    #pragma unroll
    for (int k = 0; k < K2_; ++k) f[k] = filt[n * GK2_ + g * K2_ + k];
    const float gap  = pooled[nc];
    const float ins  = inside_all[c];
    const float ins1 = ins + 1.0f;
    const float ll   = lamb_l[c];
    const float lh1  = lamb_h[c] + 1.0f;
    const float bias = -ins * gap;                  // fold (-inside*gap)

    // Column reflection.
    const int wm = (tid == 0)      ? 1      : tid - 1;
    const int wp = (tid == W_ - 1) ? W_ - 2 : tid + 1;

    #pragma unroll 4
    for (int r = 0; r < ROWS; ++r) {
        float low = f[0] * tile[r    ][wm] + f[1] * tile[r    ][tid] + f[2] * tile[r    ][wp]
                  + f[3] * tile[r + 1][wm] + f[4] * tile[r + 1][tid] + f[5] * tile[r + 1][wp]
                  + f[6] * tile[r + 2][wm] + f[7] * tile[r + 2][tid] + f[8] * tile[r + 2][wp];
        const float xc = tile[r + 1][tid];
        obase[(band * ROWS + r) * W_ + tid] = (low * ins1 + bias) * ll + xc * lh1;
    }
}

// ---------------------------------------------------------------------------
extern "C" void kernel_launch(void* const* d_in, const int* in_sizes, int n_in,
                              void* d_out, int out_size, void* d_ws, size_t ws_size,
                              hipStream_t stream) {
    const float* x        = (const float*)d_in[0];
    const float* conv_w   = (const float*)d_in[1];
    const float* bn_gamma = (const float*)d_in[2];
    const float* bn_beta  = (const float*)d_in[3];
    const float* bn_mean  = (const float*)d_in[4];
    const float* bn_var   = (const float*)d_in[5];
    const float* lamb_l   = (const float*)d_in[6];
    const float* lamb_h   = (const float*)d_in[7];
    const float* inside   = (const float*)d_in[8];
    float* out = (float*)d_out;

    float* pooled = (float*)d_ws;                   // 128 floats
    float* filt   = pooled + N_ * C_;               // 288 floats

    pool_mean_kernel<<<N_ * C_, 256, 0, stream>>>(x, pooled);
    filter_wmma_kernel<<<1, 32, 0, stream>>>(pooled, conv_w, bn_gamma, bn_beta,
                                             bn_mean, bn_var, filt);
    dynfilt_stencil_kernel<<<N_ * C_ * (H_ / ROWS), 256, 0, stream>>>(
        x, pooled, filt, lamb_l, lamb_h, inside, out);
}